// memristor_device_32796370272915
// MI455X (gfx1250) — compile-verified
//
#include <hip/hip_runtime.h>

typedef float v4f __attribute__((ext_vector_type(4)));

#define TPB 256
#define NSAMP 64
#define PAIRS (1024 * 1024)

// read-noise scale polynomial
#define A2C (-0.0058f)
#define A1C (0.0324f)
#define A0C (0.0141f)

__device__ __forceinline__ unsigned lds_raw_offset(void* p) {
  // generic -> AS3 -> 32-bit raw LDS byte address (what VDST of
  // global_load_async_to_lds_* expects; HW adds the wave's LDS_BASE)
  return (unsigned)(unsigned long long)((__attribute__((address_space(3))) char*)p);
}

__device__ __forceinline__ void async_copy_48B(const float* g, void* l) {
  // Three b128 async global->LDS transfers; INST_OFFSET is added to BOTH the
  // LDS and the global address (CDNA5 ISA, async-tensor ch. 4.4), so a single
  // address pair covers all three.
  unsigned ldsoff = lds_raw_offset(l);
  unsigned long long gaddr = (unsigned long long)g;
  asm volatile(
      "global_load_async_to_lds_b128 %0, %1, off\n\t"
      "global_load_async_to_lds_b128 %0, %1, off offset:16\n\t"
      "global_load_async_to_lds_b128 %0, %1, off offset:32"
      :
      : "v"(ldsoff), "v"(gaddr)
      : "memory");
}

__device__ __forceinline__ void wait_async_le3() {
#if __has_builtin(__builtin_amdgcn_s_wait_asynccnt)
  __builtin_amdgcn_s_wait_asynccnt(3);
#else
  asm volatile("s_wait_asynccnt 3");
#endif
  asm volatile("" ::: "memory");
}

__device__ __forceinline__ void wait_async_0() {
#if __has_builtin(__builtin_amdgcn_s_wait_asynccnt)
  __builtin_amdgcn_s_wait_asynccnt(0);
#else
  asm volatile("s_wait_asynccnt 0");
#endif
  asm volatile("" ::: "memory");
}

__global__ __launch_bounds__(TPB) void memristor_stream_kernel(
    const float* __restrict__ state,   // (1024,1024,3)
    const float* __restrict__ u,       // (64,1024,1024,3)
    float* __restrict__ out)           // [read_sum_N (64*1M) | mus (1M) | std (1M)]
{
  const int tid = threadIdx.x;
  const size_t io_base = ((size_t)blockIdx.x * TPB + (size_t)tid) * 4;

  // ---- per-thread constants from state: 12 floats = 4 (i,o) pairs x N=3 ----
  const v4f* sp = (const v4f*)(state + io_base * 3);
  v4f s0 = sp[0], s1 = sp[1], s2 = sp[2];
  float sv[12] = {s0.x, s0.y, s0.z, s0.w, s1.x, s1.y, s1.z, s1.w,
                  s2.x, s2.y, s2.z, s2.w};
  float scale[12], base[4];
#pragma unroll
  for (int k = 0; k < 12; ++k)
    scale[k] = fmaf(fmaf(A2C, sv[k], A1C), sv[k], A0C);
#pragma unroll
  for (int j = 0; j < 4; ++j)
    base[j] = sv[3 * j] + sv[3 * j + 1] + sv[3 * j + 2];

  float accA[4] = {0.f, 0.f, 0.f, 0.f};  // sum over S of noise-sum
  float accQ[4] = {0.f, 0.f, 0.f, 0.f};  // sum over S of (noise-sum)^2

  auto process = [&](const float* ub, int s) {
    float ns[4];
#pragma unroll
    for (int j = 0; j < 4; ++j) {
      float a = 0.f;
#pragma unroll
      for (int n = 0; n < 3; ++n) {
        const int k = 3 * j + n;
        float v = ub[k] - 0.5f;
        float t = fmaf(-2.0f, fabsf(v), 1.0f);   // 1 - 2|v|
        float lg = __logf(t);                    // log1p(-2|v|), <= 0
        a += copysignf(-scale[k] * lg, v);       // Laplace noise
      }
      ns[j] = a;
      accA[j] += a;
      accQ[j] = fmaf(a, a, accQ[j]);
    }
    v4f rs = {base[0] + ns[0], base[1] + ns[1], base[2] + ns[2], base[3] + ns[3]};
    __builtin_nontemporal_store(rs, (v4f*)(out + (size_t)s * PAIRS + io_base));
  };

  // Double-buffered async global->LDS pipeline. Each thread's 48B region is
  // wave-private (written and read back only by the same thread), so only
  // per-wave s_wait_asynccnt is needed -- no workgroup barriers.
  __shared__ alignas(16) float lds[2][TPB * 12];

  auto issue = [&](int s) {
    asm volatile("" ::: "memory");
    const float* g = u + (size_t)s * ((size_t)PAIRS * 3) + io_base * 3;
    async_copy_48B(g, &lds[s & 1][tid * 12]);
  };

  issue(0);
  for (int s = 0; s < NSAMP; ++s) {
    if (s + 1 < NSAMP) {
      issue(s + 1);      // prefetch next sample into the other buffer
      wait_async_le3();  // current buffer's 3 transfers complete
    } else {
      wait_async_0();
    }
    const v4f* lp = (const v4f*)&lds[s & 1][tid * 12];
    v4f u0 = lp[0], u1 = lp[1], u2 = lp[2];   // ds_load_b128 x3
    float ub[12] = {u0.x, u0.y, u0.z, u0.w, u1.x, u1.y, u1.z, u1.w,
                    u2.x, u2.y, u2.z, u2.w};
    process(ub, s);
  }

  // ---- mean and unbiased std over the S axis ----
  v4f mu, sd;
#pragma unroll
  for (int j = 0; j < 4; ++j) {
    float m = accA[j] * (1.0f / NSAMP);
    mu[j] = base[j] + m;
    float var = fmaf(-accA[j], m, accQ[j]) * (1.0f / (NSAMP - 1));
    sd[j] = __builtin_sqrtf(fmaxf(var, 0.0f));
  }
  *(v4f*)(out + (size_t)NSAMP * PAIRS + io_base) = mu;
  *(v4f*)(out + (size_t)(NSAMP + 1) * PAIRS + io_base) = sd;
}

extern "C" void kernel_launch(void* const* d_in, const int* in_sizes, int n_in,
                              void* d_out, int out_size, void* d_ws, size_t ws_size,
                              hipStream_t stream) {
  (void)in_sizes; (void)n_in; (void)d_ws; (void)ws_size; (void)out_size;
  const float* state = (const float*)d_in[0];  // (1024,1024,3) f32
  const float* u     = (const float*)d_in[1];  // (64,1024,1024,3) f32
  // d_in[2] = no_sample (==64, baked in)
  float* out = (float*)d_out;

  const int total_threads = PAIRS / 4;         // 262144
  dim3 grid(total_threads / TPB);              // 1024 blocks
  dim3 block(TPB);
  memristor_stream_kernel<<<grid, block, 0, stream>>>(state, u, out);
}